// WTA_18708877541407
// MI455X (gfx1250) — compile-verified
//
#include <hip/hip_runtime.h>
#include <hip/hip_bf16.h>

// top-k(k=64) per row of a 16384x8192 fp32 matrix, scattered back into a
// dense zero matrix. Pure bandwidth problem: 1 GiB traffic -> ~45us floor at
// 23.3 TB/s. Strategy: one 512-thread block (16 wave32) per row; async
// global->LDS staging (CDNA5 ASYNCcnt path), register-resident radix select
// (8-bit digits, 4 passes, ds_add_u32 histograms), nontemporal B128 output.

#define COLS 8192
#define TPB  512            // 16 waves of 32
#define EPT  (COLS / TPB)   // 16 elements per thread
#define V4PT (EPT / 4)      // 4 float4 chunks per thread

typedef float __attribute__((ext_vector_type(4))) v4f;
typedef int   __attribute__((ext_vector_type(4))) v4i;

#if defined(__has_builtin)
#if __has_builtin(__builtin_amdgcn_global_load_async_to_lds_b128)
#define TOPK_HAVE_ASYNC 1
#endif
#endif

__device__ __forceinline__ unsigned key_of(float f) {
  // monotonic float -> uint transform (larger float => larger key)
  unsigned u = __float_as_uint(f);
  return u ^ (unsigned)(((int)u >> 31) | (int)0x80000000);
}

__device__ __forceinline__ float val_of(unsigned k) {
  // inverse transform
  unsigned u = k ^ ((k & 0x80000000u) ? 0x80000000u : 0xFFFFFFFFu);
  return __uint_as_float(u);
}

__global__ __launch_bounds__(TPB) void topk_scatter_kernel(
    const float* __restrict__ x, const int* __restrict__ kptr,
    float* __restrict__ out) {
  __shared__ alignas(16) float lds_row[COLS];   // 32 KB staged row
  __shared__ unsigned hist[256];                // radix histogram
  __shared__ unsigned s_prefix, s_rem, s_tick;

  const int t   = threadIdx.x;
  const int row = blockIdx.x;
  const float* __restrict__ grow = x + (size_t)row * COLS;

  // ---- stage row into LDS (async global->LDS on gfx1250) ----
#if TOPK_HAVE_ASYNC
  #pragma unroll
  for (int j = 0; j < V4PT; ++j) {
    const v4f* gp = (const v4f*)grow + (t * V4PT + j);
    v4f* lp = (v4f*)lds_row + (t * V4PT + j);
    __builtin_amdgcn_global_load_async_to_lds_b128(
        (__attribute__((address_space(1))) v4i*)gp,
        (__attribute__((address_space(3))) v4i*)lp,
        0, 0);
  }
  #if __has_builtin(__builtin_amdgcn_s_wait_asynccnt)
  __builtin_amdgcn_s_wait_asynccnt(0);
  #else
  asm volatile("s_wait_asynccnt 0" ::: "memory");
  #endif
#else
  #pragma unroll
  for (int j = 0; j < V4PT; ++j) {
    ((v4f*)lds_row)[t * V4PT + j] = ((const v4f*)grow)[t * V4PT + j];
  }
#endif
  __syncthreads();

  // ---- pull this thread's 16 elements out of LDS as sortable keys ----
  unsigned key[EPT];
  #pragma unroll
  for (int j = 0; j < V4PT; ++j) {
    v4f v = ((const v4f*)lds_row)[t * V4PT + j];
    key[4 * j + 0] = key_of(v.x);
    key[4 * j + 1] = key_of(v.y);
    key[4 * j + 2] = key_of(v.z);
    key[4 * j + 3] = key_of(v.w);
  }

  int k = *kptr;
  if (k < 1) k = 1;
  if (k > COLS) k = COLS;
  if (t == 0) { s_prefix = 0u; s_rem = (unsigned)k; s_tick = 0u; }

  // ---- 4-pass MSB-first radix select (8-bit digits) for kth-largest key ----
  #pragma unroll
  for (int p = 0; p < 4; ++p) {
    const int pos = 24 - 8 * p;
    if (t < 256) hist[t] = 0u;
    __syncthreads();
    const unsigned prefix = s_prefix;
    const unsigned pmask  = p ? (0xFFFFFFFFu << ((32 - 8 * p) & 31)) : 0u;
    #pragma unroll
    for (int e = 0; e < EPT; ++e) {
      if ((key[e] & pmask) == prefix)
        atomicAdd(&hist[(key[e] >> pos) & 255u], 1u);
    }
    __syncthreads();
    if (t < 32) {
      // wave 0: suffix-sum over 256 bins (8 bins per lane + shfl Kogge-Stone)
      unsigned v[8], sfx[8];
      #pragma unroll
      for (int i = 0; i < 8; ++i) v[i] = hist[t * 8 + i];
      unsigned run = 0;
      #pragma unroll
      for (int i = 7; i >= 0; --i) { run += v[i]; sfx[i] = run; }
      unsigned incl = run;
      #pragma unroll
      for (int d = 1; d < 32; d <<= 1) {
        unsigned o = __shfl_down(incl, d, 32);
        if (t + d < 32) incl += o;
      }
      const unsigned ex  = incl - run;   // count in strictly-higher lanes
      const unsigned rem = s_rem;
      #pragma unroll
      for (int i = 0; i < 8; ++i) {
        const unsigned S   = sfx[i] + ex;  // count with digit >= this bin
        const unsigned Sab = S - v[i];     // count with digit >  this bin
        if (S >= rem && Sab < rem) {       // exactly one lane/bin hits this
          s_prefix = prefix | ((unsigned)(t * 8 + i) << pos);
          s_rem    = rem - Sab;
        }
      }
    }
    __syncthreads();
  }

  const unsigned T    = s_prefix;  // full key of the kth-largest element
  const unsigned ties = s_rem;     // how many key==T elements to keep

  // ---- dense output: value where selected, else 0; NT B128 stores ----
  float* __restrict__ orow = out + (size_t)row * COLS;
  #pragma unroll
  for (int j = 0; j < V4PT; ++j) {
    v4f o;
    #pragma unroll
    for (int c = 0; c < 4; ++c) {
      const unsigned kk = key[4 * j + c];
      bool take = kk > T;
      if (!take && kk == T) take = (atomicAdd(&s_tick, 1u) < ties);
      o[c] = take ? val_of(kk) : 0.0f;
    }
    __builtin_nontemporal_store(o, (v4f*)orow + (t * V4PT + j));
  }
}

extern "C" void kernel_launch(void* const* d_in, const int* in_sizes, int n_in,
                              void* d_out, int out_size, void* d_ws, size_t ws_size,
                              hipStream_t stream) {
  (void)n_in; (void)out_size; (void)d_ws; (void)ws_size;
  const float* x  = (const float*)d_in[0];
  const int* kptr = (const int*)d_in[1];
  float* out      = (float*)d_out;
  const int rows  = in_sizes[0] / COLS;   // 16384
  topk_scatter_kernel<<<rows, TPB, 0, stream>>>(x, kptr, out);
}